// CrossModalConceptAttention_54631984005204
// MI455X (gfx1250) — compile-verified
//
#include <hip/hip_runtime.h>
#include <hip/hip_bf16.h>

typedef float v2f __attribute__((ext_vector_type(2)));
typedef float v8f __attribute__((ext_vector_type(8)));

#define ROWS_PER_BLOCK 128
#define KPAD 44   // 42 -> padded K (multiple of 4)
#define NPAD 48   // 42 -> padded N (3 tiles of 16)
#define YSTRIDE 52

// ---------------------------------------------------------------------------
// Precompute: fold each branch's  kv_w -> Wv -> out_w  affine chain into one
// 42x42 matrix (plus identity for the residual) and one 42 bias vector.
//   attended[o] = sum_t ctx[t] * A[o][t],
//   A[o][t] = sum_i out_w[o][i] * sum_j Wv[i][j] * kv_w[j][t]
//   Wv = in_w rows [2E..3E),  bv = in_b[2E..3E)
// Stored transposed: Wt[n][k] = A[o][t] + (k==n),  n = branch_off + o, k = t.
// ---------------------------------------------------------------------------
__global__ void cmca_precompute(
    const float* inw0, const float* inb0, const float* outw0, const float* outb0,
    const float* kvw0, const float* kvb0,
    const float* inw1, const float* inb1, const float* outw1, const float* outb1,
    const float* kvw1, const float* kvb1,
    const float* inw2, const float* inb2, const float* outw2, const float* outb2,
    const float* kvw2, const float* kvb2,
    float* __restrict__ Wt, float* __restrict__ cb)
{
    const int tid = threadIdx.x;
    const int nthreads = blockDim.x;

    // zero-init (covers all padding)
    for (int i = tid; i < NPAD * KPAD; i += nthreads) Wt[i] = 0.0f;
    if (tid < NPAD) cb[tid] = 0.0f;
    __syncthreads();

    const float* IW[3] = {inw0, inw1, inw2};
    const float* IB[3] = {inb0, inb1, inb2};
    const float* OW[3] = {outw0, outw1, outw2};
    const float* OB[3] = {outb0, outb1, outb2};
    const float* KW[3] = {kvw0, kvw1, kvw2};
    const float* KB[3] = {kvb0, kvb1, kvb2};
    const int Es[3]   = {15, 14, 13};
    const int offs[3] = {0, 15, 29};

    for (int br = 0; br < 3; ++br) {
        const int E = Es[br];
        const int off = offs[br];
        const float* in_w  = IW[br];
        const float* in_b  = IB[br];
        const float* out_w = OW[br];
        const float* out_b = OB[br];
        const float* kv_w  = KW[br];
        const float* kv_b  = KB[br];

        // matrix elements
        for (int idx = tid; idx < E * 42; idx += nthreads) {
            const int o = idx / 42;
            const int t = idx % 42;
            float A = 0.0f;
            for (int i = 0; i < E; ++i) {
                float s = 0.0f;
                for (int j = 0; j < E; ++j)
                    s += in_w[(2 * E + i) * E + j] * kv_w[j * 42 + t];
                A += out_w[o * E + i] * s;
            }
            const int n = off + o;
            const int k = t;
            Wt[n * KPAD + k] = A + ((k == n) ? 1.0f : 0.0f);
        }
        // bias elements
        for (int o = tid; o < E; o += nthreads) {
            float c = out_b[o];
            for (int i = 0; i < E; ++i) {
                float tb = in_b[2 * E + i];
                for (int j = 0; j < E; ++j)
                    tb += in_w[(2 * E + i) * E + j] * kv_b[j];
                c += out_w[o * E + i] * tb;
            }
            cb[off + o] = c;
        }
    }
}

// ---------------------------------------------------------------------------
// Main fused kernel: Y = ctx @ W + c  (f32 WMMA 16x16x4), then segment-LN.
// 256 threads (8 wave32), 128 rows per block, each wave owns a 16-row tile.
// ---------------------------------------------------------------------------
__global__ __launch_bounds__(256)
void cmca_fused(const float* __restrict__ cxr,
                const float* __restrict__ ecg,
                const float* __restrict__ ehr,
                const float* __restrict__ Wt,
                const float* __restrict__ cb,
                const float* __restrict__ g_cxr, const float* __restrict__ b_cxr,
                const float* __restrict__ g_ecg, const float* __restrict__ b_ecg,
                const float* __restrict__ g_ehr, const float* __restrict__ b_ehr,
                float* __restrict__ out_cxr,
                float* __restrict__ out_ecg,
                float* __restrict__ out_ehr)
{
    __shared__ float ctxs[ROWS_PER_BLOCK * KPAD];   // 22528 B
    __shared__ float wls[NPAD * KPAD];              //  8448 B  (W transposed: [n][k])
    __shared__ float ys[ROWS_PER_BLOCK * YSTRIDE];  // 26624 B
    __shared__ float cbs[NPAD], gls[NPAD], bls[NPAD];

    const int tid = threadIdx.x;
    const long rowBase = (long)blockIdx.x * ROWS_PER_BLOCK;

    // ---- stage weights / bias / LN params into LDS ----
    for (int i = tid; i < NPAD * KPAD; i += 256) wls[i] = Wt[i];
    if (tid < NPAD) cbs[tid] = cb[tid];
    if (tid < NPAD) {
        float g = 0.0f, b = 0.0f;
        if (tid < 15)       { g = g_cxr[tid];       b = b_cxr[tid];       }
        else if (tid < 29)  { g = g_ecg[tid - 15];  b = b_ecg[tid - 15];  }
        else if (tid < 42)  { g = g_ehr[tid - 29];  b = b_ehr[tid - 29];  }
        gls[tid] = g; bls[tid] = b;
    }
    // K padding columns 42..43 = 0
    for (int r = tid; r < ROWS_PER_BLOCK; r += 256) {
        ctxs[r * KPAD + 42] = 0.0f;
        ctxs[r * KPAD + 43] = 0.0f;
    }
    // ---- coalesced load of ctx tile ----
    for (int i = tid; i < ROWS_PER_BLOCK * 15; i += 256) {
        int r = i / 15, c = i % 15;
        ctxs[r * KPAD + c] = cxr[rowBase * 15 + i];
    }
    for (int i = tid; i < ROWS_PER_BLOCK * 14; i += 256) {
        int r = i / 14, c = i % 14;
        ctxs[r * KPAD + 15 + c] = ecg[rowBase * 14 + i];
    }
    for (int i = tid; i < ROWS_PER_BLOCK * 13; i += 256) {
        int r = i / 13, c = i % 13;
        ctxs[r * KPAD + 29 + c] = ehr[rowBase * 13 + i];
    }
    __syncthreads();

    // ---- per-wave 16x48 GEMM tile via v_wmma_f32_16x16x4_f32 ----
    const int wave = tid >> 5;
    const int lane = tid & 31;
    const int l15  = lane & 15;
    const int h    = lane >> 4;       // half-wave: K offset +2 (ISA A-layout)
    const int waveRow = wave * 16;

    v8f acc0, acc1, acc2;
    {
        const float c0 = cbs[l15];
        const float c1 = cbs[16 + l15];
        const float c2 = cbs[32 + l15];
        for (int i = 0; i < 8; ++i) { acc0[i] = c0; acc1[i] = c1; acc2[i] = c2; }
    }

    const float* arow = &ctxs[(waveRow + l15) * KPAD + 2 * h];
    const float* b0p  = &wls[(0  + l15) * KPAD + 2 * h];
    const float* b1p  = &wls[(16 + l15) * KPAD + 2 * h];
    const float* b2p  = &wls[(32 + l15) * KPAD + 2 * h];

    for (int k0 = 0; k0 < KPAD; k0 += 4) {
        v2f a  = *(const v2f*)(arow + k0);   // A: row l15, K = k0 + 2h + {0,1}
        v2f b0 = *(const v2f*)(b0p + k0);    // B: col l15, K = k0 + 2h + {0,1}
        v2f b1 = *(const v2f*)(b1p + k0);
        v2f b2 = *(const v2f*)(b2p + k0);
        acc0 = __builtin_amdgcn_wmma_f32_16x16x4_f32(false, a, false, b0, (short)0, acc0, false, false);
        acc1 = __builtin_amdgcn_wmma_f32_16x16x4_f32(false, a, false, b1, (short)0, acc1, false, false);
        acc2 = __builtin_amdgcn_wmma_f32_16x16x4_f32(false, a, false, b2, (short)0, acc2, false, false);
    }

    // ---- spill y tile to LDS (C/D layout: vgpr i -> row i + 8h, lane -> col) ----
    for (int i = 0; i < 8; ++i) {
        int r = waveRow + i + 8 * h;
        ys[r * YSTRIDE + 0  + l15] = acc0[i];
        ys[r * YSTRIDE + 16 + l15] = acc1[i];
        ys[r * YSTRIDE + 32 + l15] = acc2[i];
    }
    __syncthreads();

    // ---- per-row segment LayerNorm ----
    if (tid < ROWS_PER_BLOCK) {
        float* yr = &ys[tid * YSTRIDE];
        const int segs[4] = {0, 15, 29, 42};
        for (int s = 0; s < 3; ++s) {
            const int a0 = segs[s], a1 = segs[s + 1];
            const float n = (float)(a1 - a0);
            float mu = 0.0f;
            for (int c = a0; c < a1; ++c) mu += yr[c];
            mu /= n;
            float var = 0.0f;
            for (int c = a0; c < a1; ++c) { float d = yr[c] - mu; var += d * d; }
            var /= n;
            const float inv = rsqrtf(var + 1e-5f);
            for (int c = a0; c < a1; ++c)
                yr[c] = (yr[c] - mu) * inv * gls[c] + bls[c];
        }
    }
    __syncthreads();

    // ---- coalesced stores ----
    for (int i = tid; i < ROWS_PER_BLOCK * 15; i += 256) {
        int r = i / 15, c = i % 15;
        out_cxr[rowBase * 15 + i] = ys[r * YSTRIDE + c];
    }
    for (int i = tid; i < ROWS_PER_BLOCK * 14; i += 256) {
        int r = i / 14, c = i % 14;
        out_ecg[rowBase * 14 + i] = ys[r * YSTRIDE + 15 + c];
    }
    for (int i = tid; i < ROWS_PER_BLOCK * 13; i += 256) {
        int r = i / 13, c = i % 13;
        out_ehr[rowBase * 13 + i] = ys[r * YSTRIDE + 29 + c];
    }
}

// ---------------------------------------------------------------------------
extern "C" void kernel_launch(void* const* d_in, const int* in_sizes, int n_in,
                              void* d_out, int out_size, void* d_ws, size_t ws_size,
                              hipStream_t stream) {
    (void)n_in; (void)out_size; (void)ws_size;
    const float* cxr = (const float*)d_in[0];
    const float* ecg = (const float*)d_in[1];
    const float* ehr = (const float*)d_in[2];

    // per-branch param base indices: cxr@3, ecg@11, ehr@19
    const float* inw[3]  = {(const float*)d_in[3],  (const float*)d_in[11], (const float*)d_in[19]};
    const float* inb[3]  = {(const float*)d_in[4],  (const float*)d_in[12], (const float*)d_in[20]};
    const float* outw[3] = {(const float*)d_in[5],  (const float*)d_in[13], (const float*)d_in[21]};
    const float* outb[3] = {(const float*)d_in[6],  (const float*)d_in[14], (const float*)d_in[22]};
    const float* kvw[3]  = {(const float*)d_in[7],  (const float*)d_in[15], (const float*)d_in[23]};
    const float* kvb[3]  = {(const float*)d_in[8],  (const float*)d_in[16], (const float*)d_in[24]};
    const float* lng[3]  = {(const float*)d_in[9],  (const float*)d_in[17], (const float*)d_in[25]};
    const float* lnb[3]  = {(const float*)d_in[10], (const float*)d_in[18], (const float*)d_in[26]};

    float* Wt = (float*)d_ws;          // NPAD*KPAD floats
    float* cb = Wt + NPAD * KPAD;      // NPAD floats

    cmca_precompute<<<1, 256, 0, stream>>>(
        inw[0], inb[0], outw[0], outb[0], kvw[0], kvb[0],
        inw[1], inb[1], outw[1], outb[1], kvw[1], kvb[1],
        inw[2], inb[2], outw[2], outb[2], kvw[2], kvb[2],
        Wt, cb);

    const long Brows = (long)in_sizes[0] / 15;   // 2097152
    float* out = (float*)d_out;
    float* ocxr = out;
    float* oecg = out + Brows * 15;
    float* oehr = out + Brows * 29;

    const int grid = (int)(Brows / ROWS_PER_BLOCK);
    cmca_fused<<<grid, 256, 0, stream>>>(
        cxr, ecg, ehr, Wt, cb,
        lng[0], lnb[0], lng[1], lnb[1], lng[2], lnb[2],
        ocxr, oecg, oehr);
}